// DSANSS_21852793602790
// MI455X (gfx1250) — compile-verified
//
#include <hip/hip_runtime.h>
#include <math.h>

#define DEV __device__ __forceinline__

typedef __attribute__((ext_vector_type(2))) float v2f;
typedef __attribute__((ext_vector_type(8))) float v8f;

// ---------------- constants ----------------
static constexpr int BATCH = 64;
static constexpr int NBANDC = 198;
static constexpr int HW = 81;          // 9x9 spatial
static constexpr int DMOD = 288;
static constexpr int DIN  = 576;
static constexpr int DSTT = 16;
static constexpr int DRK  = 18;
static constexpr int LSEQ = 81;
static constexpr int MSEQ = BATCH * LSEQ;   // 5184
static constexpr int D1   = 96;             // depth after conv1 (stride-2, k=7)
static constexpr int DHW  = D1 * HW;        // 7776

DEV float dev_sigmoid(float x) { return 1.f / (1.f + __expf(-x)); }
DEV float dev_silu(float x)    { return x * dev_sigmoid(x); }
DEV float dev_softplus(float x){ return (x > 20.f) ? x : log1pf(__expf(x)); }
DEV float dev_bnrelu(float x, int n, const float* g, const float* b,
                     const float* m, const float* v) {
  float s = g[n] * rsqrtf(v[n] + 1e-5f);
  float r = (x - m[n]) * s + b[n];
  return fmaxf(r, 0.f);
}

// ---------------- A-matrix loaders ----------------
struct RowMajorA {
  const float* A; int ld;
  DEV float operator()(int m, int k) const { return A[(size_t)m * ld + k]; }
};
struct RevRowsA {           // per-batch time reversal, row = (b, 80-l)
  const float* A; int ld;
  DEV float operator()(int m, int k) const {
    int b = m / LSEQ, l = m - b * LSEQ;
    return A[((size_t)(b * LSEQ + (LSEQ - 1 - l))) * ld + k];
  }
};
struct Conv7A {             // implicit im2col: X (B,24,96,81), k=(c*7+t), depth pad 3
  const float* X;
  DEV float operator()(int m, int k) const {
    int b = m / DHW; int r = m - b * DHW;
    int d = r / HW;  int hw = r - d * HW;
    int c = k / 7;   int t = k - c * 7;
    int dd = d + t - 3;
    if (dd < 0 || dd >= D1) return 0.f;
    return X[(((size_t)b * 24 + c) * D1 + dd) * HW + hw];
  }
};
struct C4A {                // X (B,24,96,81); m=(b*81+hw), k=(c*96+d)
  const float* X;
  DEV float operator()(int m, int k) const {
    int b = m / HW, hw = m - b * HW;
    int c = k / D1, d = k - c * D1;
    return X[(((size_t)b * 24 + c) * D1 + d) * HW + hw];
  }
};

// ---------------- epilogues ----------------
struct EpBnReluConv {       // store (b,n,d,hw), x = relu(bn(v+bias))
  float* O; const float* bias; const float* g; const float* bb;
  const float* mm; const float* vv;
  DEV void operator()(int m, int n, float v) const {
    float x = dev_bnrelu(v + bias[n], n, g, bb, mm, vv);
    int b = m / DHW; int r = m - b * DHW; int d = r / HW; int hw = r - d * HW;
    O[(((size_t)b * 24 + n) * D1 + d) * HW + hw] = x;
  }
};
struct EpBnReluRes {        // in-place: O[idx] = relu(bn(v+bias)) + O[idx]
  float* O; const float* bias; const float* g; const float* bb;
  const float* mm; const float* vv;
  DEV void operator()(int m, int n, float v) const {
    float x = dev_bnrelu(v + bias[n], n, g, bb, mm, vv);
    int b = m / DHW; int r = m - b * DHW; int d = r / HW; int hw = r - d * HW;
    size_t idx = (((size_t)b * 24 + n) * D1 + d) * HW + hw;
    O[idx] = x + O[idx];
  }
};
struct EpC4 {               // seq[(b*81+hw)*288 + n] = relu(bn4(v+bias))
  float* seq; const float* bias; const float* g; const float* bb;
  const float* mm; const float* vv;
  DEV void operator()(int m, int n, float v) const {
    seq[(size_t)m * DMOD + n] = dev_bnrelu(v + bias[n], n, g, bb, mm, vv);
  }
};
struct EpSplitXZ {          // xz -> xb (n<576) / z (n>=576)
  float* xb; float* z;
  DEV void operator()(int m, int n, float v) const {
    if (n < DIN) xb[(size_t)m * DIN + n] = v;
    else         z [(size_t)m * DIN + (n - DIN)] = v;
  }
};
struct EpStoreLd {
  float* C; int ld;
  DEV void operator()(int m, int n, float v) const { C[(size_t)m * ld + n] = v; }
};
struct EpCat {              // cat[(b, maybe-rev l)*576 + choff + n] = v
  float* cat; int choff; int rev;
  DEV void operator()(int m, int n, float v) const {
    int b = m / LSEQ, l = m - b * LSEQ;
    int ml = rev ? (b * LSEQ + (LSEQ - 1 - l)) : m;
    cat[(size_t)ml * (2 * DMOD) + choff + n] = v;
  }
};
struct EpComb {             // seq[m,n] = v + comb_b[n] + seq[m,n] (residual, in-place)
  float* seq; const float* cb;
  DEV void operator()(int m, int n, float v) const {
    size_t idx = (size_t)m * DMOD + n;
    seq[idx] = v + cb[n] + seq[idx];
  }
};

// ---------------- generic fp32 WMMA GEMM ----------------
// C[M,N] = A[M,K] * B[K,N], B row-major (K x N). Block tile 64(M) x NT*16(N),
// NT compile-time: no runtime predication in the WMMA loop. Both A and B are
// staged K-contiguous in LDS (B stored transposed: Bs[col][k]) so every
// fragment is one paired 2-dword LDS load straight into an aligned VGPR pair
// -- no fixup movs between the load clause and the NT back-to-back WMMAs.
// Fragment layouts per CDNA5 ISA: A 16x4 (K 0/1 lanes0-15, K 2/3 lanes16-31),
// B 4x16 (rows 0,2 | 1,3 split across lane halves), C 8 VGPRs.
template <int NT, typename AL, typename EP>
__global__ __launch_bounds__(128) void wmma_gemm_kernel(
    AL al, const float* __restrict__ Bm, EP ep, int M, int N, int K, int bn0) {
  __shared__ float As[64][33];
  __shared__ float Bs[NT * 16][33];   // transposed: [col][k]
  const int tid  = threadIdx.x;
  const int wave = tid >> 5;
  const int lane = tid & 31;
  const int bm = blockIdx.y * 64;
  const int bn = bn0 + blockIdx.x * 64;

  v8f acc[NT];
#pragma unroll
  for (int t = 0; t < NT; ++t)
#pragma unroll
    for (int j = 0; j < 8; ++j) acc[t][j] = 0.f;

  const int row   = (wave << 4) + (lane & 15);
  const int khalf = (lane >> 4) << 1;   // 0 or 2
  const int nl    = lane & 15;
  const int ncols = NT * 16;

  for (int kt = 0; kt < K; kt += 32) {
    for (int i = tid; i < 64 * 32; i += 128) {
      int r = i >> 5, kk = i & 31;
      int gm = bm + r, gk = kt + kk;
      As[r][kk] = (gm < M && gk < K) ? al(gm, gk) : 0.f;
    }
    for (int i = tid; i < 32 * ncols; i += 128) {
      int kk = i / ncols, c = i - kk * ncols;
      int gk = kt + kk, gn = bn + c;
      Bs[c][kk] = (gk < K && gn < N) ? Bm[(size_t)gk * N + gn] : 0.f;
    }
    __syncthreads();
#pragma unroll
    for (int kk = 0; kk < 32; kk += 4) {
      v2f a;
      a.x = As[row][kk + khalf];
      a.y = As[row][kk + khalf + 1];
      v2f bf[NT];
#pragma unroll
      for (int t = 0; t < NT; ++t) {
        bf[t].x = Bs[(t << 4) + nl][kk + khalf];
        bf[t].y = Bs[(t << 4) + nl][kk + khalf + 1];
      }
#pragma unroll
      for (int t = 0; t < NT; ++t)
        acc[t] = __builtin_amdgcn_wmma_f32_16x16x4_f32(
            false, a, false, bf[t], (short)0, acc[t], false, false);
    }
    __syncthreads();
  }

  const int mbase = bm + (wave << 4) + ((lane >> 4) << 3);
#pragma unroll
  for (int t = 0; t < NT; ++t) {
    int n = bn + (t << 4) + nl;
    if (n >= N) continue;
#pragma unroll
    for (int j = 0; j < 8; ++j) {
      int m = mbase + j;
      if (m < M) ep(m, n, acc[t][j]);
    }
  }
}

// Host-side dispatcher: full 64-wide blocks with NT=4, plus one edge launch
// for the N remainder with the exact compile-time tile count.
template <typename AL, typename EP>
static void launch_gemm(AL al, const float* Bm, EP ep, int M, int N, int K,
                        hipStream_t stream) {
  int mt = (M + 63) / 64;
  int nfull = N / 64;
  if (nfull > 0)
    wmma_gemm_kernel<4, AL, EP><<<dim3(nfull, mt), 128, 0, stream>>>(
        al, Bm, ep, M, N, K, 0);
  int rem = N - nfull * 64;
  if (rem > 0) {
    int bn0 = nfull * 64;
    int nt = (rem + 15) / 16;
    dim3 g(1, mt);
    switch (nt) {
      case 1: wmma_gemm_kernel<1, AL, EP><<<g, 128, 0, stream>>>(al, Bm, ep, M, N, K, bn0); break;
      case 2: wmma_gemm_kernel<2, AL, EP><<<g, 128, 0, stream>>>(al, Bm, ep, M, N, K, bn0); break;
      case 3: wmma_gemm_kernel<3, AL, EP><<<g, 128, 0, stream>>>(al, Bm, ep, M, N, K, bn0); break;
      default: wmma_gemm_kernel<4, AL, EP><<<g, 128, 0, stream>>>(al, Bm, ep, M, N, K, bn0); break;
    }
  }
}

// ---------------- small VALU kernels ----------------
__global__ __launch_bounds__(256) void transpose_kernel(
    const float* __restrict__ in, float* __restrict__ out, int R, int C) {
  int i = blockIdx.x * 256 + threadIdx.x;
  if (i >= R * C) return;
  int r = i / C, c = i - r * C;
  out[(size_t)c * R + r] = in[i];
}

// c1: (B,1,198,9,9) --k7 s2--> relu(bn1) (B,24,96,81)
__global__ __launch_bounds__(256) void c1_kernel(
    const float* __restrict__ x, const float* __restrict__ w,
    const float* __restrict__ bias, const float* g, const float* bb,
    const float* mm, const float* vv, float* __restrict__ O) {
  int i = blockIdx.x * 256 + threadIdx.x;
  if (i >= BATCH * 24 * D1 * HW) return;
  int hw = i % HW; int t1 = i / HW; int d = t1 % D1; int t2 = t1 / D1;
  int o = t2 % 24; int b = t2 / 24;
  float s = bias[o];
#pragma unroll
  for (int t = 0; t < 7; ++t)
    s += w[o * 7 + t] * x[((size_t)b * NBANDC + 2 * d + t) * HW + hw];
  O[i] = dev_bnrelu(s, o, g, bb, mm, vv);
}

// c5: full-depth conv (k=198) -> relu(bn5) (B,24,81)
__global__ __launch_bounds__(256) void c5_kernel(
    const float* __restrict__ x, const float* __restrict__ w,
    const float* __restrict__ bias, const float* g, const float* bb,
    const float* mm, const float* vv, float* __restrict__ O) {
  int i = blockIdx.x * 256 + threadIdx.x;
  if (i >= BATCH * 24 * HW) return;
  int hw = i % HW; int t1 = i / HW; int o = t1 % 24; int b = t1 / 24;
  float s = bias[o];
  for (int t = 0; t < NBANDC; ++t)
    s += w[o * NBANDC + t] * x[((size_t)b * NBANDC + t) * HW + hw];
  O[i] = dev_bnrelu(s, o, g, bb, mm, vv);
}

// c8: 1x1 conv 24->96 (+bias, no bn/relu)
__global__ __launch_bounds__(256) void c8_kernel(
    const float* __restrict__ t5, const float* __restrict__ w,
    const float* __restrict__ bias, float* __restrict__ O) {
  int i = blockIdx.x * 256 + threadIdx.x;
  if (i >= BATCH * 96 * HW) return;
  int hw = i % HW; int t1 = i / HW; int o = t1 % 96; int b = t1 / 96;
  float s = bias[o];
  for (int c = 0; c < 24; ++c)
    s += w[o * 24 + c] * t5[((size_t)b * 24 + c) * HW + hw];
  O[i] = s;
}

// c6: 3x3 spatial 24->24, relu(bn6)
__global__ __launch_bounds__(256) void c6_kernel(
    const float* __restrict__ t5, const float* __restrict__ w,
    const float* __restrict__ bias, const float* g, const float* bb,
    const float* mm, const float* vv, float* __restrict__ O) {
  int i = blockIdx.x * 256 + threadIdx.x;
  if (i >= BATCH * 24 * HW) return;
  int hw = i % HW; int t1 = i / HW; int o = t1 % 24; int b = t1 / 24;
  int h = hw / 9, wc = hw - h * 9;
  float s = bias[o];
  for (int c = 0; c < 24; ++c)
#pragma unroll
    for (int ky = 0; ky < 3; ++ky) {
      int hh = h + ky - 1; if (hh < 0 || hh >= 9) continue;
#pragma unroll
      for (int kx = 0; kx < 3; ++kx) {
        int ww = wc + kx - 1; if (ww < 0 || ww >= 9) continue;
        s += w[(o * 24 + c) * 9 + ky * 3 + kx] *
             t5[((size_t)b * 24 + c) * HW + hh * 9 + ww];
      }
    }
  O[i] = dev_bnrelu(s, o, g, bb, mm, vv);
}

// c7: 3x3 spatial 24->96, relu(bn7) + res2(t8); write seq channels 192..287
__global__ __launch_bounds__(256) void c7_kernel(
    const float* __restrict__ t6, const float* __restrict__ w,
    const float* __restrict__ bias, const float* g, const float* bb,
    const float* mm, const float* vv, const float* __restrict__ t8,
    float* __restrict__ seq) {
  int i = blockIdx.x * 256 + threadIdx.x;
  if (i >= BATCH * 96 * HW) return;
  int hw = i % HW; int t1 = i / HW; int o = t1 % 96; int b = t1 / 96;
  int h = hw / 9, wc = hw - h * 9;
  float s = bias[o];
  for (int c = 0; c < 24; ++c)
#pragma unroll
    for (int ky = 0; ky < 3; ++ky) {
      int hh = h + ky - 1; if (hh < 0 || hh >= 9) continue;
#pragma unroll
      for (int kx = 0; kx < 3; ++kx) {
        int ww = wc + kx - 1; if (ww < 0 || ww >= 9) continue;
        s += w[(o * 24 + c) * 9 + ky * 3 + kx] *
             t6[((size_t)b * 24 + c) * HW + hh * 9 + ww];
      }
    }
  float x = dev_bnrelu(s, o, g, bb, mm, vv) + t8[i];
  seq[((size_t)b * HW + hw) * DMOD + 192 + o] = x;
}

// layernorm over DM=288, one block per row
__global__ __launch_bounds__(128) void ln_kernel(
    const float* __restrict__ seq, const float* __restrict__ g,
    const float* __restrict__ b, float* __restrict__ xn) {
  __shared__ float red[128];
  int m = blockIdx.x, tid = threadIdx.x;
  const float* rowp = seq + (size_t)m * DMOD;
  float s = 0.f;
  for (int c = tid; c < DMOD; c += 128) s += rowp[c];
  red[tid] = s; __syncthreads();
  for (int st = 64; st > 0; st >>= 1) { if (tid < st) red[tid] += red[tid + st]; __syncthreads(); }
  float mu = red[0] / (float)DMOD; __syncthreads();
  float v = 0.f;
  for (int c = tid; c < DMOD; c += 128) { float d = rowp[c] - mu; v += d * d; }
  red[tid] = v; __syncthreads();
  for (int st = 64; st > 0; st >>= 1) { if (tid < st) red[tid] += red[tid + st]; __syncthreads(); }
  float rstd = rsqrtf(red[0] / (float)DMOD + 1e-5f);
  for (int c = tid; c < DMOD; c += 128)
    xn[(size_t)m * DMOD + c] = (rowp[c] - mu) * rstd * g[c] + b[c];
}

// depthwise causal conv (k=4, left pad 3) + silu
__global__ __launch_bounds__(256) void dwconv_kernel(
    const float* __restrict__ xb, const float* __restrict__ cw,
    const float* __restrict__ cb, float* __restrict__ xc) {
  int i = blockIdx.x * 256 + threadIdx.x;
  if (i >= MSEQ * DIN) return;
  int ch = i % DIN; int m = i / DIN;
  int b = m / LSEQ, l = m - b * LSEQ;
  float s = cb[ch];
#pragma unroll
  for (int t = 0; t < 4; ++t) {
    int ll = l - 3 + t;
    if (ll >= 0) s += cw[ch * 4 + t] * xb[((size_t)(b * LSEQ + ll)) * DIN + ch];
  }
  xc[i] = dev_silu(s);
}

// dt = softplus(xdbl[:, :18] @ dt_w.T + dt_b)
__global__ __launch_bounds__(256) void dt_kernel(
    const float* __restrict__ xdbl, const float* __restrict__ dtw,
    const float* __restrict__ dtb, float* __restrict__ dt) {
  int i = blockIdx.x * 256 + threadIdx.x;
  if (i >= MSEQ * DIN) return;
  int ch = i % DIN; int m = i / DIN;
  float s = dtb[ch];
#pragma unroll
  for (int r = 0; r < DRK; ++r) s += xdbl[(size_t)m * 50 + r] * dtw[ch * DRK + r];
  dt[i] = dev_softplus(s);
}

// selective scan: one thread per (b, ch); 16-state recurrence over 81 steps,
// fused with y = (scan + x*D) * silu(z)
__global__ __launch_bounds__(256) void scan_kernel(
    const float* __restrict__ dt, const float* __restrict__ xc,
    const float* __restrict__ xdbl, const float* __restrict__ z,
    const float* __restrict__ Alog, const float* __restrict__ Dp,
    float* __restrict__ y) {
  int idx = blockIdx.x * 256 + threadIdx.x;
  if (idx >= BATCH * DIN) return;
  int ch = idx % DIN; int b = idx / DIN;
  float An[DSTT], h[DSTT];
#pragma unroll
  for (int s = 0; s < DSTT; ++s) { An[s] = -__expf(Alog[ch * DSTT + s]); h[s] = 0.f; }
  float Dv = Dp[ch];
  for (int t = 0; t < LSEQ; ++t) {
    size_t m = (size_t)(b * LSEQ + t);
    float dtv = dt[m * DIN + ch];
    float xv  = xc[m * DIN + ch];
    float dx  = dtv * xv;
    const float* xd = xdbl + m * 50;
    float acc = 0.f;
#pragma unroll
    for (int s = 0; s < DSTT; ++s) {
      float Bv = xd[DRK + s];
      float Cv = xd[DRK + DSTT + s];
      h[s] = __expf(dtv * An[s]) * h[s] + dx * Bv;
      acc += h[s] * Cv;
    }
    float zv = z[m * DIN + ch];
    y[m * DIN + ch] = (acc + xv * Dv) * dev_silu(zv);
  }
}

// channel-attention pooling: avg & max over spatial per (b,c)
__global__ __launch_bounds__(256) void ca_pool_kernel(
    const float* __restrict__ seq, float* __restrict__ avg, float* __restrict__ mx) {
  int i = blockIdx.x * 256 + threadIdx.x;
  if (i >= BATCH * DMOD) return;
  int c = i % DMOD, b = i / DMOD;
  float s = 0.f, mxv = -3.4e38f;
  for (int l = 0; l < HW; ++l) {
    float v = seq[((size_t)b * HW + l) * DMOD + c];
    s += v; mxv = fmaxf(mxv, v);
  }
  avg[i] = s / (float)HW; mx[i] = mxv;
}

__global__ __launch_bounds__(256) void ca_hid_kernel(
    const float* __restrict__ avg, const float* __restrict__ mx,
    const float* __restrict__ fc1, float* __restrict__ hA, float* __restrict__ hM) {
  int i = blockIdx.x * 256 + threadIdx.x;
  if (i >= 2 * BATCH * DRK) return;
  int which = i / (BATCH * DRK);
  int r = i - which * BATCH * DRK;
  int j = r % DRK, b = r / DRK;
  const float* src = which ? mx : avg;
  float s = 0.f;
  for (int c = 0; c < DMOD; ++c) s += src[b * DMOD + c] * fc1[j * DMOD + c];
  s = fmaxf(s, 0.f);
  (which ? hM : hA)[b * DRK + j] = s;
}

__global__ __launch_bounds__(256) void ca_out_kernel(
    const float* __restrict__ hA, const float* __restrict__ hM,
    const float* __restrict__ fc2, float* __restrict__ ca) {
  int i = blockIdx.x * 256 + threadIdx.x;
  if (i >= BATCH * DMOD) return;
  int c = i % DMOD, b = i / DMOD;
  float s = 0.f;
#pragma unroll
  for (int j = 0; j < DRK; ++j)
    s += (hA[b * DRK + j] + hM[b * DRK + j]) * fc2[c * DRK + j];
  ca[i] = dev_sigmoid(s);
}

__global__ __launch_bounds__(256) void sa_pool_kernel(
    const float* __restrict__ seq, float* __restrict__ smean, float* __restrict__ smax) {
  int i = blockIdx.x * 256 + threadIdx.x;
  if (i >= BATCH * HW) return;
  float s = 0.f, mxv = -3.4e38f;
  const float* rowp = seq + (size_t)i * DMOD;
  for (int c = 0; c < DMOD; ++c) { float v = rowp[c]; s += v; mxv = fmaxf(mxv, v); }
  smean[i] = s / (float)DMOD; smax[i] = mxv;
}

// 7x7 spatial conv on (mean,max) planes (only kd=3 tap survives D=1 + pad3), sigmoid
__global__ __launch_bounds__(256) void sa_conv_kernel(
    const float* __restrict__ smean, const float* __restrict__ smax,
    const float* __restrict__ saw, float* __restrict__ sa) {
  int i = blockIdx.x * 256 + threadIdx.x;
  if (i >= BATCH * HW) return;
  int l = i % HW, b = i / HW;
  int h = l / 9, wc = l - h * 9;
  float s = 0.f;
  for (int ky = 0; ky < 7; ++ky) {
    int hh = h + ky - 3; if (hh < 0 || hh >= 9) continue;
    for (int kx = 0; kx < 7; ++kx) {
      int ww = wc + kx - 3; if (ww < 0 || ww >= 9) continue;
      int src = b * HW + hh * 9 + ww;
      s += saw[(0 * 7 + 3) * 49 + ky * 7 + kx] * smean[src];
      s += saw[(1 * 7 + 3) * 49 + ky * 7 + kx] * smax[src];
    }
  }
  sa[i] = dev_sigmoid(s);
}

// pooled[b,c] = ca[b,c] * (sum_l sa[b,l]*f[b,l,c]) / 81
__global__ __launch_bounds__(256) void pooled_kernel(
    const float* __restrict__ seq, const float* __restrict__ ca,
    const float* __restrict__ sa, float* __restrict__ pooled) {
  int i = blockIdx.x * 256 + threadIdx.x;
  if (i >= BATCH * DMOD) return;
  int c = i % DMOD, b = i / DMOD;
  float s = 0.f;
  for (int l = 0; l < HW; ++l)
    s += sa[b * HW + l] * seq[((size_t)b * HW + l) * DMOD + c];
  pooled[i] = ca[i] * s / (float)HW;
}

// heads: f, l2norm(f@h1+b), l2norm(f@h2+b), f@fc1+b, sigmoid(f@fc2+b)
__global__ __launch_bounds__(128) void heads_kernel(
    const float* __restrict__ pooled,
    const float* __restrict__ h1w, const float* __restrict__ h1b,
    const float* __restrict__ h2w, const float* __restrict__ h2b,
    const float* __restrict__ fc1w, const float* __restrict__ fc1b,
    const float* __restrict__ fc2w, const float* __restrict__ fc2b,
    float* __restrict__ out, int base) {
  __shared__ float f[DMOD];
  __shared__ float red[128];
  int b = blockIdx.x, tid = threadIdx.x;
  for (int c = tid; c < DMOD; c += 128) {
    float v = pooled[b * DMOD + c];
    f[c] = v;
    out[base + b * DMOD + c] = v;
  }
  __syncthreads();
  // h1
  float t1 = h1b[tid];
  for (int c = 0; c < DMOD; ++c) t1 += f[c] * h1w[tid * DMOD + c];
  red[tid] = t1 * t1; __syncthreads();
  for (int s = 64; s > 0; s >>= 1) { if (tid < s) red[tid] += red[tid + s]; __syncthreads(); }
  float n1 = fmaxf(sqrtf(red[0]), 1e-12f);
  out[base + 18432 + b * 128 + tid] = t1 / n1;
  __syncthreads();
  // h2
  float t2 = h2b[tid];
  for (int c = 0; c < DMOD; ++c) t2 += f[c] * h2w[tid * DMOD + c];
  red[tid] = t2 * t2; __syncthreads();
  for (int s = 64; s > 0; s >>= 1) { if (tid < s) red[tid] += red[tid + s]; __syncthreads(); }
  float n2 = fmaxf(sqrtf(red[0]), 1e-12f);
  out[base + 26624 + b * 128 + tid] = t2 / n2;
  __syncthreads();
  if (tid < 3) {
    float a = fc1b[tid];
    for (int c = 0; c < DMOD; ++c) a += f[c] * fc1w[tid * DMOD + c];
    out[base + 34816 + b * 3 + tid] = a;
  }
  if (tid == 0) {
    float a = fc2b[0];
    for (int c = 0; c < DMOD; ++c) a += f[c] * fc2w[c];
    out[base + 35008 + b] = dev_sigmoid(a);
  }
}

// ---------------- param indices (setup_inputs dict order, leaves DFS) ----------------
enum {
  IN_X = 0, IN_Y,
  P_C1W, P_C1B, P_BN1G, P_BN1B, P_BN1M, P_BN1V,
  P_C2W, P_C2B, P_BN2G, P_BN2B, P_BN2M, P_BN2V,
  P_C3W, P_C3B, P_BN3G, P_BN3B, P_BN3M, P_BN3V,
  P_C4W, P_C4B, P_BN4G, P_BN4B, P_BN4M, P_BN4V,
  P_C5W, P_C5B, P_BN5G, P_BN5B, P_BN5M, P_BN5V,
  P_C6W, P_C6B, P_BN6G, P_BN6B, P_BN6M, P_BN6V,
  P_C7W, P_C7B, P_BN7G, P_BN7B, P_BN7M, P_BN7V,
  P_C8W, P_C8B, P_LNG, P_LNB,
  PF_INW, PF_CONVW, PF_CONVB, PF_XPW, PF_DTW, PF_DTB, PF_ALOG, PF_D, PF_OUTW,
  PB_INW, PB_CONVW, PB_CONVB, PB_XPW, PB_DTW, PB_DTB, PB_ALOG, PB_D, PB_OUTW,
  P_COMBW, P_COMBB, P_CAFC1, P_CAFC2, P_SAW,
  P_H1W, P_H1B, P_H2W, P_H2B, P_FC1W, P_FC1B, P_FC2W, P_FC2B
};

extern "C" void kernel_launch(void* const* d_in, const int* in_sizes, int n_in,
                              void* d_out, int out_size, void* d_ws, size_t ws_size,
                              hipStream_t stream) {
  (void)in_sizes; (void)n_in; (void)out_size; (void)ws_size;
  auto P = [&](int i) { return (const float*)d_in[i]; };
  float* ws = (float*)d_ws;
  float* out = (float*)d_out;

  // ---- workspace layout (floats) ----
  constexpr size_t NA = (size_t)BATCH * 24 * D1 * HW;   // 11,943,936
  constexpr size_t SB = (size_t)MSEQ * DIN;             //  2,985,984
  float* bufA = ws;                 // c1 out / c3 out (in-place res)
  float* bufB = ws + NA;            // c2 out
  // mamba overlay in [0, 2*NA)  (CNN big buffers are dead by then)
  float* xb   = ws;
  float* zb   = ws + SB;
  float* xcb  = ws + 2 * SB;
  float* dtb  = ws + 3 * SB;
  float* yb   = ws + 4 * SB;
  float* cat  = ws + 5 * SB;                       // (5184, 576)
  float* xdbl = ws + 6 * SB;                       // (5184, 50)
  size_t S = 2 * NA;
  float* seq    = ws + S;             S += (size_t)MSEQ * DMOD;
  float* xn     = ws + S;             S += (size_t)MSEQ * DMOD;
  float* t5     = ws + S;             S += (size_t)BATCH * 24 * HW;
  float* t6     = ws + S;             S += (size_t)BATCH * 24 * HW;
  float* t8     = ws + S;             S += (size_t)BATCH * 96 * HW;
  float* avgb   = ws + S;             S += BATCH * DMOD;
  float* mxb    = ws + S;             S += BATCH * DMOD;
  float* hidA   = ws + S;             S += BATCH * DRK;
  float* hidM   = ws + S;             S += BATCH * DRK;
  float* cab    = ws + S;             S += BATCH * DMOD;
  float* smean  = ws + S;             S += BATCH * HW;
  float* smax   = ws + S;             S += BATCH * HW;
  float* sab    = ws + S;             S += BATCH * HW;
  float* pooled = ws + S;             S += BATCH * DMOD;
  float* inT_f  = ws + S;             S += 288 * 1152;
  float* inT_b  = ws + S;             S += 288 * 1152;
  float* xpT_f  = ws + S;             S += 576 * 50;
  float* xpT_b  = ws + S;             S += 576 * 50;
  float* outT_f = ws + S;             S += 576 * 288;
  float* outT_b = ws + S;             S += 576 * 288;
  float* combT  = ws + S;             S += 576 * 288;
  float* c2T    = ws + S;             S += 168 * 24;
  float* c3T    = ws + S;             S += 168 * 24;
  float* c4T    = ws + S;             S += 2304 * 192;

  auto launchT = [&](const float* src, float* dst, int R, int C) {
    int n = R * C;
    transpose_kernel<<<(n + 255) / 256, 256, 0, stream>>>(src, dst, R, C);
  };
  // weight pre-transposes (B operands must be K x N row-major)
  launchT(P(PF_INW), inT_f, 1152, 288);
  launchT(P(PB_INW), inT_b, 1152, 288);
  launchT(P(PF_XPW), xpT_f, 50, 576);
  launchT(P(PB_XPW), xpT_b, 50, 576);
  launchT(P(PF_OUTW), outT_f, 288, 576);
  launchT(P(PB_OUTW), outT_b, 288, 576);
  launchT(P(P_COMBW), combT, 288, 576);
  launchT(P(P_C2W), c2T, 24, 168);
  launchT(P(P_C3W), c3T, 24, 168);
  launchT(P(P_C4W), c4T, 192, 2304);

  const int MCONV = BATCH * D1 * HW;   // 497,664

  for (int br = 0; br < 2; ++br) {
    const float* xin = P(br ? IN_Y : IN_X);
    const int base = br * 35072;

    // ---- CNN branch 1 ----
    {
      int n = BATCH * 24 * D1 * HW;
      c1_kernel<<<(n + 255) / 256, 256, 0, stream>>>(
          xin, P(P_C1W), P(P_C1B), P(P_BN1G), P(P_BN1B), P(P_BN1M), P(P_BN1V), bufA);
    }
    launch_gemm(Conv7A{bufA}, c2T,
                EpBnReluConv{bufB, P(P_C2B), P(P_BN2G), P(P_BN2B), P(P_BN2M), P(P_BN2V)},
                MCONV, 24, 168, stream);
    launch_gemm(Conv7A{bufB}, c3T,
                EpBnReluRes{bufA, P(P_C3B), P(P_BN3G), P(P_BN3B), P(P_BN3M), P(P_BN3V)},
                MCONV, 24, 168, stream);
    launch_gemm(C4A{bufA}, c4T,
                EpC4{seq, P(P_C4B), P(P_BN4G), P(P_BN4B), P(P_BN4M), P(P_BN4V)},
                MSEQ, 192, 2304, stream);

    // ---- CNN branch 2 ----
    {
      int n = BATCH * 24 * HW;
      c5_kernel<<<(n + 255) / 256, 256, 0, stream>>>(
          xin, P(P_C5W), P(P_C5B), P(P_BN5G), P(P_BN5B), P(P_BN5M), P(P_BN5V), t5);
      int n8 = BATCH * 96 * HW;
      c8_kernel<<<(n8 + 255) / 256, 256, 0, stream>>>(t5, P(P_C8W), P(P_C8B), t8);
      c6_kernel<<<(n + 255) / 256, 256, 0, stream>>>(
          t5, P(P_C6W), P(P_C6B), P(P_BN6G), P(P_BN6B), P(P_BN6M), P(P_BN6V), t6);
      c7_kernel<<<(n8 + 255) / 256, 256, 0, stream>>>(
          t6, P(P_C7W), P(P_C7B), P(P_BN7G), P(P_BN7B), P(P_BN7M), P(P_BN7V), t8, seq);
    }

    // ---- BiSSM ----
    ln_kernel<<<MSEQ, 128, 0, stream>>>(seq, P(P_LNG), P(P_LNB), xn);

    for (int dir = 0; dir < 2; ++dir) {
      const float* inT  = dir ? inT_b  : inT_f;
      const float* xpT  = dir ? xpT_b  : xpT_f;
      const float* outT = dir ? outT_b : outT_f;
      const float* cw   = P(dir ? PB_CONVW : PF_CONVW);
      const float* cb   = P(dir ? PB_CONVB : PF_CONVB);
      const float* dtw  = P(dir ? PB_DTW   : PF_DTW);
      const float* dtbi = P(dir ? PB_DTB   : PF_DTB);
      const float* alog = P(dir ? PB_ALOG  : PF_ALOG);
      const float* Dp   = P(dir ? PB_D     : PF_D);

      if (dir == 0)
        launch_gemm(RowMajorA{xn, DMOD}, inT, EpSplitXZ{xb, zb},
                    MSEQ, 1152, DMOD, stream);
      else
        launch_gemm(RevRowsA{xn, DMOD}, inT, EpSplitXZ{xb, zb},
                    MSEQ, 1152, DMOD, stream);

      {
        int n = MSEQ * DIN;
        dwconv_kernel<<<(n + 255) / 256, 256, 0, stream>>>(xb, cw, cb, xcb);
      }
      launch_gemm(RowMajorA{xcb, DIN}, xpT, EpStoreLd{xdbl, 50},
                  MSEQ, 50, DIN, stream);
      {
        int n = MSEQ * DIN;
        dt_kernel<<<(n + 255) / 256, 256, 0, stream>>>(xdbl, dtw, dtbi, dtb);
      }
      {
        int n = BATCH * DIN;
        scan_kernel<<<(n + 255) / 256, 256, 0, stream>>>(
            dtb, xcb, xdbl, zb, alog, Dp, yb);
      }
      launch_gemm(RowMajorA{yb, DIN}, outT, EpCat{cat, dir ? DMOD : 0, dir},
                  MSEQ, DMOD, DIN, stream);
    }
    launch_gemm(RowMajorA{cat, 2 * DMOD}, combT, EpComb{seq, P(P_COMBB)},
                MSEQ, DMOD, 2 * DMOD, stream);

    // ---- CBAM + pool ----
    {
      int n = BATCH * DMOD;
      ca_pool_kernel<<<(n + 255) / 256, 256, 0, stream>>>(seq, avgb, mxb);
      int nh = 2 * BATCH * DRK;
      ca_hid_kernel<<<(nh + 255) / 256, 256, 0, stream>>>(avgb, mxb, P(P_CAFC1), hidA, hidM);
      ca_out_kernel<<<(n + 255) / 256, 256, 0, stream>>>(hidA, hidM, P(P_CAFC2), cab);
      int ns = BATCH * HW;
      sa_pool_kernel<<<(ns + 255) / 256, 256, 0, stream>>>(seq, smean, smax);
      sa_conv_kernel<<<(ns + 255) / 256, 256, 0, stream>>>(smean, smax, P(P_SAW), sab);
      pooled_kernel<<<(n + 255) / 256, 256, 0, stream>>>(seq, cab, sab, pooled);
    }

    // ---- heads ----
    heads_kernel<<<BATCH, 128, 0, stream>>>(
        pooled, P(P_H1W), P(P_H1B), P(P_H2W), P(P_H2B),
        P(P_FC1W), P(P_FC1B), P(P_FC2W), P(P_FC2B), out, base);
  }
}